// EBEPModelWrapper_5514738008156
// MI455X (gfx1250) — compile-verified
//
#include <hip/hip_runtime.h>
#include <hip/hip_bf16.h>

typedef __attribute__((ext_vector_type(2))) float v2f;
typedef __attribute__((ext_vector_type(8))) float v8f;

#if __has_builtin(__builtin_amdgcn_global_load_async_to_lds_b32)
#define HAS_ASYNC_LDS 1
#else
#define HAS_ASYNC_LDS 0
#endif

__device__ __forceinline__ void wait_async0() {
#if __has_builtin(__builtin_amdgcn_s_wait_asynccnt)
  __builtin_amdgcn_s_wait_asynccnt(0);
#else
  asm volatile("s_wait_asynccnt 0" ::: "memory");
#endif
}

// GEMM modes: C[M][N] = sum_k A[M][K] * B[K][N], B gathered on the fly (implicit im2col)
enum {
  MODE_CONV1 = 0,   // a1 = conv1(x): M=256(co), K=16,   N=25088 (b,14,14)
  MODE_CONV2,       // conv2(s1):     M=64 (co), K=4096, N=6272  (b,7,7)
  MODE_CONVT2,      // conv2^T(s2):   M=256(ci), K=1024, N=25088 (b,14,14)
  MODE_CONV3,       // conv3(s2):     M=64 (co), K=576,  N=6272  (b,7,7)
  MODE_CONVT3,      // conv3^T(s3):   M=64 (ci), K=576,  N=6272  (b,7,7)
  MODE_FC_A,        // s3f @ W4^T:    M=256(j),  K=3136, N=128 (b)
  MODE_FC_B         // s4 @ W4:       M=3136(i), K=256,  N=128 (b)
};

template <int MODE> struct GT;
template <> struct GT<MODE_CONV1>  { static constexpr int M = 256,  K = 16,   N = 25088; };
template <> struct GT<MODE_CONV2>  { static constexpr int M = 64,   K = 4096, N = 6272;  };
template <> struct GT<MODE_CONVT2> { static constexpr int M = 256,  K = 1024, N = 25088; };
template <> struct GT<MODE_CONV3>  { static constexpr int M = 64,   K = 576,  N = 6272;  };
template <> struct GT<MODE_CONVT3> { static constexpr int M = 64,   K = 576,  N = 6272;  };
template <> struct GT<MODE_FC_A>   { static constexpr int M = 256,  K = 3136, N = 128;   };
template <> struct GT<MODE_FC_B>   { static constexpr int M = 3136, K = 256,  N = 128;   };

template <int MODE>
__device__ __forceinline__ float gatherB(const float* __restrict__ S, int k, int n) {
  if (k >= GT<MODE>::K) return 0.f;
  if constexpr (MODE == MODE_CONV1) {
    int b = n / 196, p = n % 196, oh = p / 14, ow = p % 14;
    int kh = k >> 2, kw = k & 3;
    int ih = 2 * oh - 1 + kh, iw = 2 * ow - 1 + kw;
    return (ih >= 0 && ih < 28 && iw >= 0 && iw < 28) ? S[b * 784 + ih * 28 + iw] : 0.f;
  } else if constexpr (MODE == MODE_CONV2) {
    int b = n / 49, p = n % 49, oh = p / 7, ow = p % 7;
    int ci = k >> 4, r = k & 15, kh = r >> 2, kw = r & 3;
    int ih = 2 * oh - 1 + kh, iw = 2 * ow - 1 + kw;
    return (ih >= 0 && ih < 14 && iw >= 0 && iw < 14)
               ? S[((b * 256 + ci) * 196) + ih * 14 + iw] : 0.f;
  } else if constexpr (MODE == MODE_CONVT2) {
    int b = n / 196, p = n % 196, ih = p / 14, iw = p % 14;
    int co = k >> 4, r = k & 15, kh = r >> 2, kw = r & 3;
    int th = ih + 1 - kh, tw = iw + 1 - kw;
    bool ok = (th >= 0) && (tw >= 0) && ((th & 1) == 0) && ((tw & 1) == 0) &&
              ((th >> 1) < 7) && ((tw >> 1) < 7);
    return ok ? S[((b * 64 + co) * 49) + (th >> 1) * 7 + (tw >> 1)] : 0.f;
  } else if constexpr (MODE == MODE_CONV3) {
    int b = n / 49, p = n % 49, oh = p / 7, ow = p % 7;
    int ci = k / 9, r = k % 9, kh = r / 3, kw = r % 3;
    int ih = oh - 1 + kh, iw = ow - 1 + kw;
    return (ih >= 0 && ih < 7 && iw >= 0 && iw < 7)
               ? S[((b * 64 + ci) * 49) + ih * 7 + iw] : 0.f;
  } else if constexpr (MODE == MODE_CONVT3) {
    int b = n / 49, p = n % 49, ih = p / 7, iw = p % 7;
    int co = k / 9, r = k % 9, kh = r / 3, kw = r % 3;
    int oh = ih + 1 - kh, ow = iw + 1 - kw;
    return (oh >= 0 && oh < 7 && ow >= 0 && ow < 7)
               ? S[((b * 64 + co) * 49) + oh * 7 + ow] : 0.f;
  } else if constexpr (MODE == MODE_FC_A) {
    return S[n * 3136 + k];
  } else {                    // MODE_FC_B
    return S[n * 256 + k];
  }
}

// Implicit GEMM on V_WMMA_F32_16X16X4_F32.
// Block = 256 thr (8 waves). Block tile 64(M) x 128(N), Kc = 32.
// Wave (wm = w&3, wn = w>>2) owns a 16x64 strip -> 4 accumulators.
// LDS is K-contiguous with pad 34 so every 16x4 / 4x16 fragment is one
// aligned ds_load_b64 into an even VGPR pair, bank-conflict-free.
template <int MODE>
__global__ __launch_bounds__(256) void gemm_wmma(const float* __restrict__ A,
                                                 const float* __restrict__ Bsrc,
                                                 float* __restrict__ C) {
  constexpr int K = GT<MODE>::K, N = GT<MODE>::N;
  __shared__ float Alds[64][34];    // [m][k]
  __shared__ float Blds[128][34];   // [n][k]
  const int bm = blockIdx.y * 64;
  const int bn = blockIdx.x * 128;
  const int tid = threadIdx.x;
  const int lane = tid & 31;
  const int wave = tid >> 5;
  const int wm = wave & 3;
  const int wn = wave >> 2;

  v8f acc0 = {0.f, 0.f, 0.f, 0.f, 0.f, 0.f, 0.f, 0.f};
  v8f acc1 = acc0, acc2 = acc0, acc3 = acc0;

  for (int k0 = 0; k0 < K; k0 += 32) {
    __syncthreads();
    // ---- Stage A (regular copy): async DMA global -> LDS where available ----
#pragma unroll
    for (int i = 0; i < 8; ++i) {
      int idx = tid + (i << 8);
      int m = idx >> 5, k = idx & 31;       // k-fastest: coalesced global reads
#if HAS_ASYNC_LDS
      if (k0 + k < K) {
        __builtin_amdgcn_global_load_async_to_lds_b32(
            (int*)&A[(bm + m) * K + k0 + k], (int*)&Alds[m][k], 0, 0);
      } else {
        Alds[m][k] = 0.f;
      }
#else
      Alds[m][k] = (k0 + k < K) ? A[(bm + m) * K + k0 + k] : 0.f;
#endif
    }
    // ---- Stage B (implicit im2col gather) overlaps the async DMA ----
#pragma unroll
    for (int i = 0; i < 16; ++i) {
      int idx = tid + (i << 8);
      int k = idx >> 7, n = idx & 127;      // n-fastest: mostly coalesced gather
      Blds[n][k] = gatherB<MODE>(Bsrc, k0 + k, bn + n);
    }
#if HAS_ASYNC_LDS
    wait_async0();
#endif
    __syncthreads();

    // A frag (16x4 f32): lanes 0-15 hold K={0,1}, lanes 16-31 hold K={2,3}.
    const int mrow = wm * 16 + (lane & 15);
    const int ksel = (lane & 16) ? 2 : 0;
    const int nb = wn * 64 + (lane & 15);
#pragma unroll
    for (int kk = 0; kk < 32; kk += 4) {
      v2f a  = *(const v2f*)&Alds[mrow][kk + ksel];
      v2f b0 = *(const v2f*)&Blds[nb][kk + ksel];
      v2f b1 = *(const v2f*)&Blds[nb + 16][kk + ksel];
      v2f b2 = *(const v2f*)&Blds[nb + 32][kk + ksel];
      v2f b3 = *(const v2f*)&Blds[nb + 48][kk + ksel];
      acc0 = __builtin_amdgcn_wmma_f32_16x16x4_f32(false, a, false, b0, (short)0,
                                                   acc0, false, false);
      acc1 = __builtin_amdgcn_wmma_f32_16x16x4_f32(false, a, false, b1, (short)0,
                                                   acc1, false, false);
      acc2 = __builtin_amdgcn_wmma_f32_16x16x4_f32(false, a, false, b2, (short)0,
                                                   acc2, false, false);
      acc3 = __builtin_amdgcn_wmma_f32_16x16x4_f32(false, a, false, b3, (short)0,
                                                   acc3, false, false);
    }
  }

  // C/D layout: VGPR v, lanes 0-15 -> M=v, N=lane; lanes 16-31 -> M=v+8, N=lane-16.
  const int rbase = bm + wm * 16 + ((lane >> 4) << 3);
  const int cbase = bn + wn * 64 + (lane & 15);
#pragma unroll
  for (int v = 0; v < 8; ++v) {
    C[(rbase + v) * N + cbase]      = acc0[v];
    C[(rbase + v) * N + cbase + 16] = acc1[v];
    C[(rbase + v) * N + cbase + 32] = acc2[v];
    C[(rbase + v) * N + cbase + 48] = acc3[v];
  }
}

// ---------- weight pre-pack kernels ----------
__global__ __launch_bounds__(256) void pack_w2t(const float* __restrict__ w2,
                                                float* __restrict__ w2t) {
  int i = blockIdx.x * 256 + threadIdx.x;        // [ci][co*16+r], 256*1024
  if (i >= 262144) return;
  int r = i & 15, co = (i >> 4) & 63, ci = i >> 10;
  w2t[i] = w2[co * 4096 + ci * 16 + r];
}
__global__ __launch_bounds__(256) void pack_w3t(const float* __restrict__ w3,
                                                float* __restrict__ w3t) {
  int i = blockIdx.x * 256 + threadIdx.x;        // [ci][co*9+r], 64*576
  if (i >= 36864) return;
  int r = i % 9, co = (i / 9) % 64, ci = i / 576;
  w3t[i] = w3[co * 576 + ci * 9 + r];
}
__global__ __launch_bounds__(256) void pack_w4t(const float* __restrict__ w4,
                                                float* __restrict__ w4t) {
  int i = blockIdx.x * 256 + threadIdx.x;        // [i][j], 3136*256
  if (i >= 802816) return;
  int j = i & 255, m = i >> 8;
  w4t[i] = w4[j * 3136 + m];
}
__global__ __launch_bounds__(256) void add_bias_a1(float* __restrict__ a1,
                                                   const float* __restrict__ b1) {
  int i = blockIdx.x * 256 + threadIdx.x;        // [ci][25088]
  a1[i] += b1[i / 25088];
}

// ---------- state update kernels (synchronous fixed-point step) ----------
__global__ __launch_bounds__(256) void update_s1(const float* __restrict__ so,
                                                 const float* __restrict__ a1,
                                                 const float* __restrict__ g1,
                                                 float* __restrict__ sn) {
  int i = blockIdx.x * 256 + threadIdx.x;        // (b,ci,p), 6,422,528
  int p = i % 196, t = i / 196, ci = t & 255, b = t >> 8;
  int gi = ci * 25088 + b * 196 + p;
  sn[i] = 0.5f * so[i] + 0.5f * (a1[gi] + g1[gi]);
}
__global__ __launch_bounds__(256) void update_s2(const float* __restrict__ so,
                                                 const float* __restrict__ ga,
                                                 const float* __restrict__ gb,
                                                 const float* __restrict__ bias,
                                                 float* __restrict__ sn) {
  int i = blockIdx.x * 256 + threadIdx.x;        // (b,co,p), 401,408
  int p = i % 49, t = i / 49, co = t & 63, b = t >> 6;
  int gi = co * 6272 + b * 49 + p;
  sn[i] = 0.5f * so[i] + 0.5f * (ga[gi] + bias[co] + gb[gi]);
}
__global__ __launch_bounds__(256) void update_s3(const float* __restrict__ so,
                                                 const float* __restrict__ ga,
                                                 const float* __restrict__ gfc,
                                                 const float* __restrict__ bias,
                                                 float* __restrict__ sn) {
  int i = blockIdx.x * 256 + threadIdx.x;        // (b,co,p), 401,408
  int p = i % 49, t = i / 49, co = t & 63, b = t >> 6;
  int gi = co * 6272 + b * 49 + p;
  float fc = gfc[(co * 49 + p) * 128 + b];       // G3B layout [3136][128]
  sn[i] = 0.5f * so[i] + 0.5f * (ga[gi] + bias[co] + fc);
}
__global__ __launch_bounds__(256) void update_s4(const float* __restrict__ so,
                                                 const float* __restrict__ g4,
                                                 const float* __restrict__ bias,
                                                 float* __restrict__ sn) {
  int i = blockIdx.x * 256 + threadIdx.x;        // (b,j), 32,768
  int j = i & 255, b = i >> 8;
  sn[i] = 0.5f * so[i] + 0.5f * (g4[j * 128 + b] + bias[j]);
}

// ---------- reductions ----------
__global__ __launch_bounds__(256) void phix_kernel(const float* __restrict__ x,
                                                   const float* __restrict__ vb,
                                                   float* __restrict__ out) {
  __shared__ float red[256];
  int b = blockIdx.x, tid = threadIdx.x;
  float s = 0.f;
  for (int i = tid; i < 784; i += 256) s += vb[i] * x[b * 784 + i];
  red[tid] = s;
  __syncthreads();
  for (int k = 128; k > 0; k >>= 1) {
    if (tid < k) red[tid] += red[tid + k];
    __syncthreads();
  }
  if (tid == 0) out[b] = red[0];
}

__global__ __launch_bounds__(256) void energy_kernel(
    const float* __restrict__ s1, const float* __restrict__ s2,
    const float* __restrict__ s3, const float* __restrict__ s4,
    const float* __restrict__ a1, const float* __restrict__ g2a,
    const float* __restrict__ g3a, const float* __restrict__ g4,
    const float* __restrict__ b2, const float* __restrict__ b3,
    const float* __restrict__ b4, const float* __restrict__ phix,
    float* __restrict__ out) {
  __shared__ float rq[256], rd[256];
  int b = blockIdx.x, tid = threadIdx.x;
  float q = 0.f, d = 0.f;
  for (int idx = tid; idx < 256 * 196; idx += 256) {           // s1
    int ci = idx / 196, p = idx % 196;
    float v = s1[b * 50176 + idx];
    q += v * v;
    d += v * a1[ci * 25088 + b * 196 + p];
  }
  for (int idx = tid; idx < 64 * 49; idx += 256) {             // s2
    int co = idx / 49, p = idx % 49;
    float v = s2[b * 3136 + idx];
    q += v * v;
    d += v * (g2a[co * 6272 + b * 49 + p] + b2[co]);
  }
  for (int idx = tid; idx < 64 * 49; idx += 256) {             // s3
    int co = idx / 49, p = idx % 49;
    float v = s3[b * 3136 + idx];
    q += v * v;
    d += v * (g3a[co * 6272 + b * 49 + p] + b3[co]);
  }
  {                                                            // s4 (j = tid)
    float v = s4[b * 256 + tid];
    q += v * v;
    d += v * (g4[tid * 128 + b] + b4[tid]);
  }
  rq[tid] = q; rd[tid] = d;
  __syncthreads();
  for (int k = 128; k > 0; k >>= 1) {
    if (tid < k) { rq[tid] += rq[tid + k]; rd[tid] += rd[tid + k]; }
    __syncthreads();
  }
  if (tid == 0) out[b] = 0.5f * rq[0] - (phix[b] + rd[0]);
}

extern "C" void kernel_launch(void* const* d_in, const int* in_sizes, int n_in,
                              void* d_out, int out_size, void* d_ws, size_t ws_size,
                              hipStream_t stream) {
  const float* x   = (const float*)d_in[0];
  const float* vb  = (const float*)d_in[1];
  const float* w1w = (const float*)d_in[2];
  const float* w1b = (const float*)d_in[3];
  const float* w2w = (const float*)d_in[4];
  const float* w2b = (const float*)d_in[5];
  const float* w3w = (const float*)d_in[6];
  const float* w3b = (const float*)d_in[7];
  const float* w4w = (const float*)d_in[8];
  const float* w4b = (const float*)d_in[9];
  (void)in_sizes; (void)n_in; (void)out_size; (void)ws_size;

  float* ws = (float*)d_ws;
  float* A1   = ws; ws += 256 * 25088;          // conv1 pre-activation [ci][b*196+p]
  float* S1[2]; S1[0] = ws; ws += 6422528; S1[1] = ws; ws += 6422528;
  float* S2[2]; S2[0] = ws; ws += 401408;  S2[1] = ws; ws += 401408;
  float* S3[2]; S3[0] = ws; ws += 401408;  S3[1] = ws; ws += 401408;
  float* S4[2]; S4[0] = ws; ws += 32768;   S4[1] = ws; ws += 32768;
  float* G1  = ws; ws += 256 * 25088;           // convT2 out
  float* G2A = ws; ws += 64 * 6272;             // conv2 out
  float* G2B = ws; ws += 64 * 6272;             // convT3 out
  float* G3A = ws; ws += 64 * 6272;             // conv3 out
  float* G3B = ws; ws += 3136 * 128;            // s4@W4 out
  float* G4  = ws; ws += 256 * 128;             // s3f@W4^T out
  float* W2T = ws; ws += 262144;
  float* W3T = ws; ws += 36864;
  float* W4T = ws; ws += 802816;
  float* PHIX = ws; ws += 128;

  // ---- one-time (per call) setup ----
  pack_w2t<<<262144 / 256, 256, 0, stream>>>(w2w, W2T);
  pack_w3t<<<(36864 + 255) / 256, 256, 0, stream>>>(w3w, W3T);
  pack_w4t<<<802816 / 256, 256, 0, stream>>>(w4w, W4T);
  gemm_wmma<MODE_CONV1><<<dim3(196, 4), 256, 0, stream>>>(w1w, x, A1);
  add_bias_a1<<<(256 * 25088) / 256, 256, 0, stream>>>(A1, w1b);
  phix_kernel<<<128, 256, 0, stream>>>(x, vb, PHIX);
  (void)hipMemsetAsync(S1[0], 0, 6422528 * sizeof(float), stream);
  (void)hipMemsetAsync(S2[0], 0, 401408 * sizeof(float), stream);
  (void)hipMemsetAsync(S3[0], 0, 401408 * sizeof(float), stream);
  (void)hipMemsetAsync(S4[0], 0, 32768 * sizeof(float), stream);

  // ---- T = 50 synchronous fixed-point iterations (ping-pong buffers) ----
  for (int t = 0; t < 50; ++t) {
    int o = t & 1, n = o ^ 1;
    gemm_wmma<MODE_CONVT2><<<dim3(196, 4), 256, 0, stream>>>(W2T, S2[o], G1);
    gemm_wmma<MODE_CONV2><<<dim3(49, 1), 256, 0, stream>>>(w2w, S1[o], G2A);
    gemm_wmma<MODE_CONVT3><<<dim3(49, 1), 256, 0, stream>>>(W3T, S3[o], G2B);
    gemm_wmma<MODE_CONV3><<<dim3(49, 1), 256, 0, stream>>>(w3w, S2[o], G3A);
    gemm_wmma<MODE_FC_B><<<dim3(1, 49), 256, 0, stream>>>(W4T, S4[o], G3B);
    gemm_wmma<MODE_FC_A><<<dim3(1, 4), 256, 0, stream>>>(w4w, S3[o], G4);
    update_s1<<<25088, 256, 0, stream>>>(S1[o], A1, G1, S1[n]);
    update_s2<<<1568, 256, 0, stream>>>(S2[o], G2A, G2B, w2b, S2[n]);
    update_s3<<<1568, 256, 0, stream>>>(S3[o], G3A, G3B, w3b, S3[n]);
    update_s4<<<128, 256, 0, stream>>>(S4[o], G4, w4b, S4[n]);
  }

  // ---- energy at the final state (T even -> final state in buffer 0) ----
  gemm_wmma<MODE_CONV2><<<dim3(49, 1), 256, 0, stream>>>(w2w, S1[0], G2A);
  gemm_wmma<MODE_CONV3><<<dim3(49, 1), 256, 0, stream>>>(w3w, S2[0], G3A);
  gemm_wmma<MODE_FC_A><<<dim3(1, 4), 256, 0, stream>>>(w4w, S3[0], G4);
  energy_kernel<<<128, 256, 0, stream>>>(S1[0], S2[0], S3[0], S4[0], A1, G2A, G3A,
                                         G4, w2b, w3b, w4b, PHIX, (float*)d_out);
}